// ZBL_84189948936817
// MI455X (gfx1250) — compile-verified
//
#include <hip/hip_runtime.h>

typedef __attribute__((ext_vector_type(4))) float vf4;
typedef __attribute__((ext_vector_type(4))) int   vi4;
typedef __attribute__((ext_vector_type(2))) float vf2;

// ---- constants ---------------------------------------------------------
#define L2E   1.4426950408889634f
#define ZBL_C1 0.02817f
#define ZBL_C2 0.28022f
#define ZBL_C3 0.50986f
#define ZBL_C4 0.18175f
#define ZBL_D1 (-0.20162f * L2E)   // pre-scaled for exp2
#define ZBL_D2 (-0.4029f  * L2E)
#define ZBL_D3 (-0.94229f * L2E)
#define ZBL_D4 (-3.1998f  * L2E)
#define ZBL_QQ 7.1998225f          // 14.399645 * 0.5
#define INV_A0 (1.0f / 0.4685f)
#define INV_RMAX (1.0f / 6.0f)
#define PZBL  0.23f

// ---- per-edge compute: cutoff * QQ * Zi*Zj/r * psi, atomically scattered
__device__ __forceinline__ void zbl_accum(float rv, float rr,
                                          float zi, float pi,
                                          float zj, float pj,
                                          int dst, float* __restrict__ out) {
  float x   = (pi + pj) * rv * INV_A0;
  float psi = ZBL_C1 * __builtin_amdgcn_exp2f(ZBL_D1 * x)
            + ZBL_C2 * __builtin_amdgcn_exp2f(ZBL_D2 * x)
            + ZBL_C3 * __builtin_amdgcn_exp2f(ZBL_D3 * x)
            + ZBL_C4 * __builtin_amdgcn_exp2f(ZBL_D4 * x);
  float eng = ZBL_QQ * zi * zj * psi / rv;
  float rr2 = rr * rr;
  float rr3 = rr2 * rr;
  float rr6 = rr3 * rr3;
  // 1 - 28 rr^6 + 48 rr^7 - 21 rr^8
  float cut = fmaf(rr6, fmaf(rr, fmaf(rr, -21.0f, 48.0f), -28.0f), 1.0f);
  unsafeAtomicAdd(&out[dst], cut * eng);   // global_atomic_add_f32, no return
}

// ---- main edge kernel: exactly nquads full quads, no tail path ----------
// 4 edges/thread, NT b128 streaming loads of r / edge_index.
__global__ void zbl_edges_v4(const float* __restrict__ r,
                             const int*   __restrict__ ei,
                             const int*   __restrict__ ej,
                             const vf2*   __restrict__ tbl,
                             float* __restrict__ out, int nquads) {
  int t = blockIdx.x * blockDim.x + threadIdx.x;
  if (t >= nquads) return;
  int base = t * 4;
  vf4 r4 = __builtin_nontemporal_load((const vf4*)(r  + base));
  vi4 i4 = __builtin_nontemporal_load((const vi4*)(ei + base));
  vi4 j4 = __builtin_nontemporal_load((const vi4*)(ej + base));
#pragma unroll
  for (int k = 0; k < 4; ++k) {
    float rv = r4[k];
    float rr = rv * INV_RMAX;
    if (rr >= 1.0f) continue;            // cutoff==0: skip gathers+atomic
    int i = i4[k], j = j4[k];
    vf2 a = tbl[i], b = tbl[j];
    zbl_accum(rv, rr, a.x, a.y, b.x, b.y, i, out);
  }
}

// ---- setup: per-atom {Z, Z^0.23} table (L2-resident, one b64 gather/endpoint)
__global__ void zbl_atom_table(const float* __restrict__ Z,
                               const int*   __restrict__ types,
                               vf2* __restrict__ tbl, int n_atoms) {
  int a = blockIdx.x * blockDim.x + threadIdx.x;
  if (a < n_atoms) {
    float z = Z[types[a]];
    vf2 v; v.x = z; v.y = powf(z, PZBL);
    tbl[a] = v;
  }
}

// ---- scalar edge kernel (tail / odd E, still uses table) ----------------
__global__ void zbl_edges_s(const float* __restrict__ r,
                            const int*   __restrict__ ei,
                            const int*   __restrict__ ej,
                            const vf2*   __restrict__ tbl,
                            float* __restrict__ out, int E) {
  int e = blockIdx.x * blockDim.x + threadIdx.x;
  if (e >= E) return;
  float rv = r[e];
  float rr = rv * INV_RMAX;
  if (rr >= 1.0f) return;
  int i = ei[e], j = ej[e];
  vf2 a = tbl[i], b = tbl[j];
  zbl_accum(rv, rr, a.x, a.y, b.x, b.y, i, out);
}

// ---- fallback if workspace too small: pow inline per edge ---------------
__global__ void zbl_edges_direct(const float* __restrict__ r,
                                 const int*   __restrict__ ei,
                                 const int*   __restrict__ ej,
                                 const float* __restrict__ Z,
                                 const int*   __restrict__ types,
                                 float* __restrict__ out, int E) {
  int e = blockIdx.x * blockDim.x + threadIdx.x;
  if (e >= E) return;
  float rv = r[e];
  float rr = rv * INV_RMAX;
  if (rr >= 1.0f) return;
  int i = ei[e], j = ej[e];
  float zi = Z[types[i]], zj = Z[types[j]];
  zbl_accum(rv, rr, zi, __powf(zi, PZBL), zj, __powf(zj, PZBL), i, out);
}

extern "C" void kernel_launch(void* const* d_in, const int* in_sizes, int n_in,
                              void* d_out, int out_size, void* d_ws, size_t ws_size,
                              hipStream_t stream) {
  const float* Z     = (const float*)d_in[0];
  const float* r     = (const float*)d_in[1];
  const float* pae   = (const float*)d_in[2];   // per_atom_energy
  const int*   types = (const int*)d_in[3];
  const int*   eidx  = (const int*)d_in[4];     // (2, E) row-major
  const int E       = in_sizes[1];
  const int n_atoms = in_sizes[2];
  const int* ei = eidx;
  const int* ej = eidx + E;
  float* out = (float*)d_out;

  // out starts as per_atom_energy; edges atomically accumulate on top.
  hipMemcpyAsync(out, pae, (size_t)n_atoms * sizeof(float),
                 hipMemcpyDeviceToDevice, stream);

  const bool have_ws = ws_size >= (size_t)n_atoms * sizeof(vf2);
  if (have_ws) {
    vf2* tbl = (vf2*)d_ws;
    zbl_atom_table<<<(n_atoms + 255) / 256, 256, 0, stream>>>(Z, types, tbl, n_atoms);
    const int nquads = E >> 2;          // full 4-edge quads
    const int tail   = E - (nquads << 2);
    if (nquads > 0) {
      zbl_edges_v4<<<(nquads + 255) / 256, 256, 0, stream>>>(r, ei, ej, tbl, out, nquads);
    }
    if (tail > 0) {
      const int off = nquads << 2;
      zbl_edges_s<<<1, 256, 0, stream>>>(r + off, ei + off, ej + off, tbl, out, tail);
    }
  } else {
    zbl_edges_direct<<<(E + 255) / 256, 256, 0, stream>>>(r, ei, ej, Z, types, out, E);
  }
}